// InfoNCELoss_2946347565103
// MI455X (gfx1250) — compile-verified
//
#include <hip/hip_runtime.h>
#include <math.h>

// InfoNCE loss, fused f32 GEMM (WMMA f32 16x16x4) + exp/mask/row-reduce epilogue.
// N=4096, E=1024, cols=2N=8192: 68.7 GFLOP f32 GEMM, 32MB data (L2-resident),
// bound by v_wmma_f32_16x16x4_f32 throughput. Each wave computes a 32x32 output
// tile (2x2 WMMA tiles -> 4 independent accumulator chains, ds:wmma = 1:1).
// LDS row stride 68 dwords (bank advance 4 mod 64) -> conflict-free b64 reads.
// K loop is double-buffered: next chunk streamed global->LDS with the CDNA5
// async DMA path (global_load_async_to_lds_b128 / ASYNCcnt) while WMMAs run.

typedef __attribute__((ext_vector_type(2))) float v2f;
typedef __attribute__((ext_vector_type(8))) float v8f;
typedef __attribute__((ext_vector_type(4))) int   v4i;
typedef __attribute__((address_space(1))) v4i     gv4i;   // global-AS int4
typedef __attribute__((address_space(3))) v4i     lv4i;   // LDS-AS int4

#define TEMP_INV   (1.0f / 0.07f)
#define EPS_F      1e-8f

#define M_TILE 64          // rows of embs1 per block (2 M-waves x 32)
#define N_TILE 128         // sim columns per step   (4 N-waves x 32)
#define KC     64          // K chunk staged in LDS
#define LDA    (KC + 4)    // 68 dwords: bank advance 4/row on 64-bank LDS
#define CSPLIT 8           // column splits across grid.y

__device__ __forceinline__ void async_copy_b128(const float* gp, float* lp) {
#if __has_builtin(__builtin_amdgcn_global_load_async_to_lds_b128)
    // param0: v4i* in global AS; param1: v4i* in LDS AS (low 32 bits of a flat
    // LDS pointer are the LDS byte offset). Integer casts avoid AS-cast issues.
    __builtin_amdgcn_global_load_async_to_lds_b128(
        (gv4i*)(size_t)gp,
        (lv4i*)(unsigned)(size_t)lp,
        0 /*offset*/, 0 /*cpol*/);
#else
    *(float4*)lp = *(const float4*)gp;   // sync fallback
#endif
}

__device__ __forceinline__ void wait_async0() {
#if __has_builtin(__builtin_amdgcn_s_wait_asynccnt)
    __builtin_amdgcn_s_wait_asynccnt(0);
#else
    asm volatile("s_wait_asynccnt 0x0" ::: "memory");
#endif
}

__global__ __launch_bounds__(256)
void infonce_gemm_kernel(const float* __restrict__ embs1,
                         const float* __restrict__ embs2,
                         const int*   __restrict__ srcs1,
                         const int*   __restrict__ srcs2,
                         float* __restrict__ negPartial,   // [N][CSPLIT]
                         float* __restrict__ posOut,       // [N]
                         int N, int E)
{
    __shared__ __align__(16) float As[2][M_TILE][LDA];   // A: embs1 rows (M x K)
    __shared__ __align__(16) float Bs[2][N_TILE][LDA];   // B: embs rows  (N x K)
    __shared__ float negPart[M_TILE][4];                 // per-row partial per N-wave
    __shared__ int   s1s[M_TILE];
    __shared__ int   sjs[N_TILE];

    const int t    = threadIdx.x;
    const int lane = t & 31;
    const int wave = t >> 5;
    const int wm   = wave & 1;     // M-wave 0..1  (32 rows each)
    const int wn   = wave >> 1;    // N-wave 0..3  (32 cols each)
    const int mL   = lane & 15;    // M (A) / N (B,C) within 16
    const int kh   = lane >> 4;    // lane-half: K pair (A/B) or M+8 (C)

    const int rowBase  = blockIdx.x * M_TILE;
    const int colChunk = (2 * N) / CSPLIT;          // 1024 columns per block
    const int colStart = blockIdx.y * colChunk;

    if (t < M_TILE) s1s[t] = srcs1[rowBase + t];

    // per-thread staging coordinates (fixed across chunks)
    const int aRow = t >> 4;
    const int aC4  = (t & 15) << 2;

    // stage one K-chunk into buffer `buf` (async, no wait here)
    auto stageA = [&](int buf, int kk) {
#pragma unroll
        for (int it = 0; it < (M_TILE * KC / 4) / 256; ++it) {   // 4 iters
            int row = aRow + it * 16;
            async_copy_b128(embs1 + (size_t)(rowBase + row) * E + kk + aC4,
                            &As[buf][row][aC4]);
        }
    };
    auto stageB = [&](int buf, int c0, int kk) {
#pragma unroll
        for (int it = 0; it < (N_TILE * KC / 4) / 256; ++it) {   // 8 iters
            int row = aRow + it * 16;
            int j   = c0 + row;
            const float* src = (j < N) ? (embs1 + (size_t)j * E)
                                       : (embs2 + (size_t)(j - N) * E);
            async_copy_b128(src + kk + aC4, &Bs[buf][row][aC4]);
        }
    };

    // per-lane running neg sums: 2 M-subtiles x 8 rows (C layout)
    float negAcc[2][8];
#pragma unroll
    for (int mi = 0; mi < 2; ++mi)
#pragma unroll
        for (int r = 0; r < 8; ++r) negAcc[mi][r] = 0.0f;

    for (int c0 = colStart; c0 < colStart + colChunk; c0 += N_TILE) {
        __syncthreads();   // previous epilogue done reading sjs; LDS bufs free
        if (t < N_TILE) {
            int j = c0 + t;
            sjs[t] = (j < N) ? srcs1[j] : srcs2[j - N];
        }

        v8f acc[2][2];
#pragma unroll
        for (int mi = 0; mi < 2; ++mi)
#pragma unroll
            for (int ni = 0; ni < 2; ++ni)
                acc[mi][ni] = (v8f){0.f,0.f,0.f,0.f,0.f,0.f,0.f,0.f};

        // pipeline prologue: stage chunk 0 into buffer 0
        stageA(0, 0);
        stageB(0, c0, 0);
        wait_async0();
        __syncthreads();

        int cur = 0;
        for (int kk = 0; kk < E; kk += KC) {
            const int nxt = cur ^ 1;
            // kick off next chunk's DMA while we compute this one
            if (kk + KC < E) {
                stageA(nxt, kk + KC);
                stageB(nxt, c0, kk + KC);
            }

            // 16 K-steps x 4 WMMAs (4 independent accumulator chains)
#pragma unroll
            for (int k4 = 0; k4 < KC; k4 += 4) {
                const int ko = k4 + 2 * kh;
                v2f a0 = *(const v2f*)&As[cur][wm * 32 +      mL][ko];
                v2f a1 = *(const v2f*)&As[cur][wm * 32 + 16 + mL][ko];
                v2f b0 = *(const v2f*)&Bs[cur][wn * 32 +      mL][ko];
                v2f b1 = *(const v2f*)&Bs[cur][wn * 32 + 16 + mL][ko];
                acc[0][0] = __builtin_amdgcn_wmma_f32_16x16x4_f32(
                                false, a0, false, b0, (short)0, acc[0][0], false, false);
                acc[0][1] = __builtin_amdgcn_wmma_f32_16x16x4_f32(
                                false, a0, false, b1, (short)0, acc[0][1], false, false);
                acc[1][0] = __builtin_amdgcn_wmma_f32_16x16x4_f32(
                                false, a1, false, b0, (short)0, acc[1][0], false, false);
                acc[1][1] = __builtin_amdgcn_wmma_f32_16x16x4_f32(
                                false, a1, false, b1, (short)0, acc[1][1], false, false);
            }

            wait_async0();     // next buffer's DMA landed (per-wave)
            __syncthreads();   // ...for all waves; also fences this buf's reads
            cur = nxt;
        }

        // Epilogue: scale, exp, mask, pos capture, neg accumulate (registers)
#pragma unroll
        for (int mi = 0; mi < 2; ++mi) {
#pragma unroll
            for (int ni = 0; ni < 2; ++ni) {
#pragma unroll
                for (int r = 0; r < 8; ++r) {
                    int m_local = wm * 32 + mi * 16 + r + 8 * kh;  // C: M = r (+8 upper)
                    int i = rowBase + m_local;
                    int j = c0 + wn * 32 + ni * 16 + mL;          // C: N = lane%16
                    float e = __expf(acc[mi][ni][r] * TEMP_INV);
                    bool isPos = (j == i + N);
                    if (isPos) posOut[i] = e;                     // unique (i) across grid
                    int  s1 = s1s[m_local];
                    int  sj = sjs[wn * 32 + ni * 16 + mL];
                    bool masked = (j == i) || isPos || (sj == s1);
                    negAcc[mi][r] += masked ? 0.0f : e;
                }
            }
        }
    }

    // deterministic reduction: butterfly over 16 lanes sharing a row, then
    // combine the 4 N-waves via LDS in fixed order, one partial per (row, by)
    __syncthreads();
#pragma unroll
    for (int mi = 0; mi < 2; ++mi) {
#pragma unroll
        for (int r = 0; r < 8; ++r) {
            float v = negAcc[mi][r];
            v += __shfl_xor(v, 1);
            v += __shfl_xor(v, 2);
            v += __shfl_xor(v, 4);
            v += __shfl_xor(v, 8);
            if (mL == 0) negPart[wm * 32 + mi * 16 + r + 8 * kh][wn] = v;
        }
    }
    __syncthreads();
    if (t < M_TILE) {
        negPartial[(size_t)(rowBase + t) * CSPLIT + blockIdx.y] =
            ((negPart[t][0] + negPart[t][1]) + negPart[t][2]) + negPart[t][3];
    }
}

__global__ __launch_bounds__(256)
void infonce_finalize_kernel(const float* __restrict__ negPartial,
                             const float* __restrict__ pos,
                             const int*   __restrict__ srcs1,
                             const int*   __restrict__ srcs2,
                             float* __restrict__ out, int N)
{
    __shared__ float ssum[256];
    __shared__ int   scnt[256];
    const int t = threadIdx.x;
    float sum = 0.0f;
    int   cnt = 0;
    for (int i = t; i < N; i += 256) {
        float ng = 0.0f;
#pragma unroll
        for (int c = 0; c < CSPLIT; ++c) ng += negPartial[(size_t)i * CSPLIT + c];
        if (srcs1[i] == srcs2[i]) {
            float p = pos[i];
            sum += -logf(p / (p + ng + EPS_F));
            cnt += 1;
        }
    }
    ssum[t] = sum; scnt[t] = cnt;
    __syncthreads();
    for (int s = 128; s > 0; s >>= 1) {
        if (t < s) { ssum[t] += ssum[t + s]; scnt[t] += scnt[t + s]; }
        __syncthreads();
    }
    if (t == 0) out[0] = ssum[0] / (float)scnt[0];
}

extern "C" void kernel_launch(void* const* d_in, const int* in_sizes, int n_in,
                              void* d_out, int out_size, void* d_ws, size_t ws_size,
                              hipStream_t stream)
{
    const float* embs1 = (const float*)d_in[0];
    const float* embs2 = (const float*)d_in[1];
    const int*   srcs1 = (const int*)d_in[2];
    const int*   srcs2 = (const int*)d_in[3];

    const int N = in_sizes[2];            // 4096
    const int E = in_sizes[0] / N;        // 1024

    float* negPartial = (float*)d_ws;                       // N*CSPLIT floats
    float* pos        = negPartial + (size_t)N * CSPLIT;    // N floats
    // every slot of negPartial/pos is written each launch -> no zero-init needed

    dim3 grid(N / M_TILE, CSPLIT);
    infonce_gemm_kernel<<<grid, dim3(256), 0, stream>>>(
        embs1, embs2, srcs1, srcs2, negPartial, pos, N, E);

    infonce_finalize_kernel<<<1, dim3(256), 0, stream>>>(
        negPartial, pos, srcs1, srcs2, (float*)d_out, N);
}